// Encoder_49520972923532
// MI455X (gfx1250) — compile-verified
//
#include <hip/hip_runtime.h>

typedef __attribute__((ext_vector_type(2))) float v2f;
typedef __attribute__((ext_vector_type(8))) float v8f;

#define F 128
#define BN_EPS 1e-5f

__device__ __forceinline__ void atomicAddF(float* p, float v) {
  // Lowers to global_atomic_add_f32 (no return) — no CAS loop.
  __hip_atomic_fetch_add(p, v, __ATOMIC_RELAXED, __HIP_MEMORY_SCOPE_AGENT);
}

__global__ void zero_f32(float* __restrict__ p, int n) {
  int i = blockIdx.x * blockDim.x + threadIdx.x;
  int stride = gridDim.x * blockDim.x;
  for (; i < n; i += stride) p[i] = 0.f;
}

// One wave per edge: lane l carries float4 = h[src][4l..4l+3]; 4 fp32 atomics to agg[dst].
// h and agg (51 MB each) are L2-resident (192 MB), so the random traffic stays on-chip.
__global__ void scatter_edges(const float* __restrict__ h,
                              const int* __restrict__ src,
                              const int* __restrict__ dst,
                              float* __restrict__ agg, int E) {
  int gid  = blockIdx.x * blockDim.x + threadIdx.x;
  int lane = gid & 31;
  int warp = gid >> 5;
  int nwarp = (gridDim.x * blockDim.x) >> 5;
  for (int e = warp; e < E; e += nwarp) {
    int s = src[e], d = dst[e];
    float4 v = ((const float4*)(h + (size_t)s * F))[lane];
    float* a = agg + (size_t)d * F + lane * 4;
    atomicAddF(a + 0, v.x);
    atomicAddF(a + 1, v.y);
    atomicAddF(a + 2, v.z);
    atomicAddF(a + 3, v.w);
  }
}

// 256 threads = 8 waves. Block handles 16 node rows; wave w owns output cols [16w,16w+16).
// relu((z@W1)+b1)@W2+b2, relu, store y; per-feature sum/sum^2 atomics for BN.
// y may alias h (per-node dependence only; block reads its rows into LDS before writing).
__global__ __launch_bounds__(256) void gin_mlp(
    const float* __restrict__ h, const float* __restrict__ agg,
    const float* __restrict__ W1, const float* __restrict__ b1,
    const float* __restrict__ W2, const float* __restrict__ b2,
    float* __restrict__ y, float* __restrict__ ssum, float* __restrict__ ssq,
    int N) {
  __shared__ float zs[16][F + 4];   // pad 4 -> conflict-free A-frag reads
  __shared__ float ts[16][F + 4];
  const int tid  = threadIdx.x;
  const int row0 = blockIdx.x * 16;

  for (int i = tid; i < 16 * F; i += 256) {
    int r = i >> 7, c = i & (F - 1);
    int n = row0 + r;
    zs[r][c] = (n < N) ? h[(size_t)n * F + c] + agg[(size_t)n * F + c] : 0.f;
  }
  __syncthreads();

  const int wave  = tid >> 5;
  const int lane  = tid & 31;
  const int nloc  = lane & 15;            // M for A-frag, N-within-tile for B/C
  const int koff  = (lane >> 4) << 1;     // lanes 16-31 carry K+2,K+3
  const int ncol  = wave * 16 + nloc;     // global output column
  const int mbase = (lane < 16) ? 0 : 8;  // C layout: vgpr v -> row v+mbase

  // ---- GEMM1: zs @ W1 ----
  v8f acc = {};
#pragma unroll 8
  for (int k0 = 0; k0 < F; k0 += 4) {
    v2f a, b;
    a.x = zs[nloc][k0 + koff];
    a.y = zs[nloc][k0 + koff + 1];
    b.x = W1[(k0 + koff) * F + ncol];
    b.y = W1[(k0 + koff + 1) * F + ncol];
    acc = __builtin_amdgcn_wmma_f32_16x16x4_f32(false, a, false, b,
                                                (short)0, acc, false, false);
  }
  const float bias1 = b1[ncol];
#pragma unroll
  for (int v = 0; v < 8; ++v)
    ts[mbase + v][ncol] = fmaxf(acc[v] + bias1, 0.f);
  __syncthreads();

  // ---- GEMM2: ts @ W2 ----
  v8f acc2 = {};
#pragma unroll 8
  for (int k0 = 0; k0 < F; k0 += 4) {
    v2f a, b;
    a.x = ts[nloc][k0 + koff];
    a.y = ts[nloc][k0 + koff + 1];
    b.x = W2[(k0 + koff) * F + ncol];
    b.y = W2[(k0 + koff + 1) * F + ncol];
    acc2 = __builtin_amdgcn_wmma_f32_16x16x4_f32(false, a, false, b,
                                                 (short)0, acc2, false, false);
  }
  const float bias2 = b2[ncol];
  float s = 0.f, sq = 0.f;
#pragma unroll
  for (int v = 0; v < 8; ++v) {
    int m = row0 + mbase + v;
    if (m < N) {
      float val = fmaxf(acc2[v] + bias2, 0.f);
      y[(size_t)m * F + ncol] = val;
      s  += val;
      sq += val * val;
    }
  }
  atomicAddF(&ssum[ncol], s);
  atomicAddF(&ssq[ncol], sq);
}

__global__ void bn_finalize(const float* __restrict__ ssum,
                            const float* __restrict__ ssq,
                            float* __restrict__ mean, float* __restrict__ rstd, int N) {
  int c = threadIdx.x;
  if (c < F) {
    float m = ssum[c] / (float)N;
    float v = ssq[c] / (float)N - m * m;
    mean[c] = m;
    rstd[c] = rsqrtf(fmaxf(v, 0.f) + BN_EPS);
  }
}

// Normalize in place (becomes next layer's h) + global add-pool into out[g, layer*F + c].
__global__ void norm_pool(float* __restrict__ y, const float* __restrict__ mean,
                          const float* __restrict__ rstd, const float* __restrict__ gamma,
                          const float* __restrict__ beta, const int* __restrict__ batch,
                          float* __restrict__ out, int layer, int N, int L) {
  int idx = blockIdx.x * blockDim.x + threadIdx.x;
  int total = N * F;
  int stride = gridDim.x * blockDim.x;
  for (; idx < total; idx += stride) {
    int n = idx >> 7;
    int c = idx & (F - 1);
    float z = (y[idx] - mean[c]) * rstd[c] * gamma[c] + beta[c];
    y[idx] = z;
    int g = batch[n];
    atomicAddF(&out[(size_t)g * (L * F) + layer * F + c], z);
  }
}

extern "C" void kernel_launch(void* const* d_in, const int* in_sizes, int n_in,
                              void* d_out, int out_size, void* d_ws, size_t ws_size,
                              hipStream_t stream) {
  const float* x     = (const float*)d_in[0];
  const int*   ei    = (const int*)d_in[1];
  const int*   batch = (const int*)d_in[2];
  const float* W1    = (const float*)d_in[3];
  const float* b1    = (const float*)d_in[4];
  const float* W2    = (const float*)d_in[5];
  const float* b2    = (const float*)d_in[6];
  const float* gamma = (const float*)d_in[7];
  const float* beta  = (const float*)d_in[8];

  const int N = in_sizes[0] / F;          // 100000
  const int E = in_sizes[1] / 2;          // 1600000
  const int L = in_sizes[3] / (F * F);    // 3
  const int* src = ei;
  const int* dst = ei + E;

  float* ws   = (float*)d_ws;
  float* agg  = ws;                         // N*F
  float* ybuf = agg + (size_t)N * F;        // N*F (in-place h for layers >= 1)
  float* ssum = ybuf + (size_t)N * F;       // F
  float* ssq  = ssum + F;                   // F
  float* mean = ssq + F;                    // F
  float* rstd = mean + F;                   // F

  float* out = (float*)d_out;
  zero_f32<<<(out_size + 255) / 256, 256, 0, stream>>>(out, out_size);

  const int aggN = N * F;
  for (int i = 0; i < L; ++i) {
    const float* h = (i == 0) ? x : ybuf;

    zero_f32<<<(aggN + 255) / 256, 256, 0, stream>>>(agg, aggN);
    zero_f32<<<1, 256, 0, stream>>>(ssum, 2 * F);

    const int warpsPerBlock = 256 / 32;
    int sblocks = (E + warpsPerBlock - 1) / warpsPerBlock;
    scatter_edges<<<sblocks, 256, 0, stream>>>(h, src, dst, agg, E);

    gin_mlp<<<(N + 15) / 16, 256, 0, stream>>>(
        h, agg, W1 + (size_t)i * F * F, b1 + (size_t)i * F,
        W2 + (size_t)i * F * F, b2 + (size_t)i * F, ybuf, ssum, ssq, N);

    bn_finalize<<<1, F, 0, stream>>>(ssum, ssq, mean, rstd, N);

    norm_pool<<<(N * F + 255) / 256, 256, 0, stream>>>(
        ybuf, mean, rstd, gamma + (size_t)i * F, beta + (size_t)i * F,
        batch, out, i, N, L);
  }
}